// MultiHeadAttentionLayer_55499567399184
// MI455X (gfx1250) — compile-verified
//
#include <hip/hip_runtime.h>
#include <hip/hip_bf16.h>
#include <math.h>

#define NUM_HEADS 8
#define OUT_DIM   32
#define IN_DIM    256
#define HD        256          // NUM_HEADS * OUT_DIM
#define ROWS_PB   32           // rows per block in GEMM
#define LDS_PITCH (IN_DIM + 4) // +4 floats: shifts banks by 4/row -> conflict-free

typedef float v2f __attribute__((ext_vector_type(2)));
typedef float v8f __attribute__((ext_vector_type(8)));
typedef int   v4i __attribute__((ext_vector_type(4)));

#if __has_builtin(__builtin_amdgcn_global_load_async_to_lds_b128)
#define HAVE_ASYNC_LDS 1
// per hipcc diagnostic: param0 = int4 addrspace(1)*, param1 = int4 addrspace(3)*
typedef __attribute__((address_space(1))) v4i* gas_v4i;
typedef __attribute__((address_space(3))) v4i* las_v4i;
#else
#define HAVE_ASYNC_LDS 0
#endif

__device__ __forceinline__ void atomAddF(float* p, float v) {
  unsafeAtomicAdd(p, v);   // hardware global_atomic_add_f32
}

__global__ void zero_f32(float* __restrict__ p, long n) {
  long i = (long)blockIdx.x * blockDim.x + threadIdx.x;
  long stride = (long)gridDim.x * blockDim.x;
  for (; i < n; i += stride) p[i] = 0.0f;
}

// ---------------------------------------------------------------------------
// Phase 1: QKV projection.  out[n, o] = sum_k h[n,k] * W[o,k] + b[o]
//
// Block = 256 threads (8 waves), computes 32 rows x 256 cols of one matrix
// (blockIdx.y selects Q/K/V). The 32x256 h tile is staged into LDS with
// gfx1250 async global->LDS copies (ASYNCcnt + s_wait_asynccnt), then each
// wave computes a 16-row x 64-col register block: per K-step of 4, ONE A
// fragment from LDS feeds FOUR v_wmma_f32_16x16x4_f32 against four B
// fragments -> 5 loads per 4 WMMAs instead of 8.
//
// Fragment layouts per CDNA5 ISA 7.12.2 (wave32):
//   A 16x4 f32 : lane (lo,hi) holds row M=lo, v0=K(2hi), v1=K(2hi+1)  (v2f)
//   B 4x16 f32 : lane holds col N=lo, v0=K(2hi), v1=K(2hi+1)          (v2f)
//   C/D 16x16  : VGPR r, lane -> element (M = r + 8*hi, N = lo)
// EXEC must be all ones for WMMA: loads clamped, only stores guarded.
// ---------------------------------------------------------------------------
__global__ __launch_bounds__(256)
void qkv_proj_wmma(const float* __restrict__ h, int N,
                   const float* __restrict__ Wq, const float* __restrict__ bq, float* __restrict__ outQ,
                   const float* __restrict__ Wk, const float* __restrict__ bk, float* __restrict__ outK,
                   const float* __restrict__ Wv, const float* __restrict__ bv, float* __restrict__ outV) {
  __shared__ float hTile[ROWS_PB * LDS_PITCH];   // 32 rows, padded pitch (~33 KB)

  const float* W;  const float* bias;  float* out;
  if (blockIdx.y == 0)      { W = Wq; bias = bq; out = outQ; }
  else if (blockIdx.y == 1) { W = Wk; bias = bk; out = outK; }
  else                      { W = Wv; bias = bv; out = outV; }

  const int tid  = threadIdx.x;
  const int wave = tid >> 5;
  const int lane = tid & 31;
  const int lo   = lane & 15;
  const int hi   = lane >> 4;
  const int row0 = blockIdx.x * ROWS_PB;

  // ---- stage h tile: 32 rows x 64 float4 = 2048 xfers, 8 per thread -------
  for (int i = tid; i < ROWS_PB * (IN_DIM / 4); i += 256) {
    const int r  = i >> 6;            // row within tile
    const int c4 = i & 63;            // float4 column
    int row = row0 + r;
    if (row > N - 1) row = N - 1;     // clamp (harmless duplicate stage)
#if HAVE_ASYNC_LDS
    __builtin_amdgcn_global_load_async_to_lds_b128(
        (gas_v4i)(h + (long)row * IN_DIM + c4 * 4),
        (las_v4i)&hTile[r * LDS_PITCH + c4 * 4],
        /*offset=*/0, /*cpol=*/0);
#else
    *(float4*)&hTile[r * LDS_PITCH + c4 * 4] =
        *(const float4*)(h + (long)row * IN_DIM + c4 * 4);
#endif
  }
#if HAVE_ASYNC_LDS
  asm volatile("s_wait_asynccnt 0" ::: "memory");
#endif
  __syncthreads();

  // ---- per-wave register block: 16 rows x 4 col-tiles ---------------------
  const int rowHalf = (wave >> 2) & 1;          // waves 0-3: rows 0-15, 4-7: 16-31
  const int colBase = (wave & 3) * 64;          // 4 col tiles of 16

  const float* __restrict__ wr0 = W + (long)(colBase +  0 + lo) * IN_DIM;
  const float* __restrict__ wr1 = W + (long)(colBase + 16 + lo) * IN_DIM;
  const float* __restrict__ wr2 = W + (long)(colBase + 32 + lo) * IN_DIM;
  const float* __restrict__ wr3 = W + (long)(colBase + 48 + lo) * IN_DIM;

  v8f acc0, acc1, acc2, acc3;
  {
    const float b0 = bias[colBase +  0 + lo];
    const float b1 = bias[colBase + 16 + lo];
    const float b2 = bias[colBase + 32 + lo];
    const float b3 = bias[colBase + 48 + lo];
#pragma unroll
    for (int r = 0; r < 8; ++r) { acc0[r] = b0; acc1[r] = b1; acc2[r] = b2; acc3[r] = b3; }
  }

  const float* aRow = &hTile[(rowHalf * 16 + lo) * LDS_PITCH];

#pragma unroll 2
  for (int k0 = 0; k0 < IN_DIM; k0 += 4) {
    const int ko = k0 + 2 * hi;
    const v2f a  = *(const v2f*)(aRow + ko);     // ds_load_b64, conflict-free
    const v2f b0 = *(const v2f*)(wr0 + ko);
    const v2f b1 = *(const v2f*)(wr1 + ko);
    const v2f b2 = *(const v2f*)(wr2 + ko);
    const v2f b3 = *(const v2f*)(wr3 + ko);
    acc0 = __builtin_amdgcn_wmma_f32_16x16x4_f32(false, a, false, b0, (short)0, acc0, false, false);
    acc1 = __builtin_amdgcn_wmma_f32_16x16x4_f32(false, a, false, b1, (short)0, acc1, false, false);
    acc2 = __builtin_amdgcn_wmma_f32_16x16x4_f32(false, a, false, b2, (short)0, acc2, false, false);
    acc3 = __builtin_amdgcn_wmma_f32_16x16x4_f32(false, a, false, b3, (short)0, acc3, false, false);
  }

#pragma unroll
  for (int r = 0; r < 8; ++r) {
    const int row = row0 + rowHalf * 16 + r + 8 * hi;
    if (row < N) {
      float* o = out + (long)row * HD + colBase + lo;
      o[ 0] = acc0[r];
      o[16] = acc1[r];
      o[32] = acc2[r];
      o[48] = acc3[r];
    }
  }
}

// ---------------------------------------------------------------------------
// Phase 2: per-edge scores + segment-sum denominator.
// 32 lanes per edge: lane = (head<<2)|part; two coalesced float4 loads per
// lane cover the 1KB K/Q rows, then 2-step shfl_xor reduction per head group.
// ---------------------------------------------------------------------------
__global__ void edge_scores(const float* __restrict__ Q, const float* __restrict__ K,
                            const int* __restrict__ src, const int* __restrict__ dst,
                            float* __restrict__ score, float* __restrict__ z, int E) {
  const int e = blockIdx.x * 8 + (threadIdx.x >> 5);
  if (e >= E) return;                    // uniform per wave
  const int lane = threadIdx.x & 31;
  const int head = lane >> 2;
  const int part = lane & 3;

  const int s = src[e];
  const int d = dst[e];
  const float4* kp = (const float4*)(K + (long)s * HD + head * OUT_DIM + part * 8);
  const float4* qp = (const float4*)(Q + (long)d * HD + head * OUT_DIM + part * 8);
  const float4 k0 = kp[0], k1 = kp[1];
  const float4 q0 = qp[0], q1 = qp[1];

  float sum = k0.x * q0.x + k0.y * q0.y + k0.z * q0.z + k0.w * q0.w
            + k1.x * q1.x + k1.y * q1.y + k1.z * q1.z + k1.w * q1.w;
  sum += __shfl_xor(sum, 1, 32);
  sum += __shfl_xor(sum, 2, 32);

  if (part == 0) {
    const float scl = 0.17677669529663687f;          // 1/sqrt(32)
    float v = fminf(fmaxf(sum * scl, -5.0f), 5.0f);
    const float sc = __expf(v);
    score[(long)e * NUM_HEADS + head] = sc;
    atomAddF(&z[(long)d * NUM_HEADS + head], sc);
  }
}

// ---------------------------------------------------------------------------
// Phase 3: wV[dst] += (score/z[dst]) * V[src]. One edge per 256-thread block,
// thread = (head, d) element; coalesced V row read, hardware f32 atomics out.
// ---------------------------------------------------------------------------
__global__ void aggregate(const float* __restrict__ V,
                          const float* __restrict__ score, const float* __restrict__ z,
                          const int* __restrict__ src, const int* __restrict__ dst,
                          float* __restrict__ out, int E) {
  const int e = blockIdx.x;
  const int t = threadIdx.x;
  const int head = t >> 5;
  const int dcol = t & 31;

  const int s = src[e];
  const int d = dst[e];
  const float alpha = score[(long)e * NUM_HEADS + head] / z[(long)d * NUM_HEADS + head];
  const float v = V[(long)s * HD + head * OUT_DIM + dcol];
  atomAddF(&out[(long)d * HD + head * OUT_DIM + dcol], alpha * v);
}

extern "C" void kernel_launch(void* const* d_in, const int* in_sizes, int n_in,
                              void* d_out, int out_size, void* d_ws, size_t ws_size,
                              hipStream_t stream) {
  const float* h  = (const float*)d_in[0];
  const float* Wq = (const float*)d_in[1];
  const float* bq = (const float*)d_in[2];
  const float* Wk = (const float*)d_in[3];
  const float* bk = (const float*)d_in[4];
  const float* Wv = (const float*)d_in[5];
  const float* bv = (const float*)d_in[6];
  const int*  src = (const int*)d_in[7];
  const int*  dst = (const int*)d_in[8];
  float* out = (float*)d_out;

  const int N = in_sizes[0] / IN_DIM;    // 50000
  const int E = in_sizes[7];             // 800000

  // workspace layout (floats): Q | K | V | z | score   (~181 MB total)
  float* Q     = (float*)d_ws;
  float* K     = Q + (size_t)N * HD;
  float* V     = K + (size_t)N * HD;
  float* z     = V + (size_t)N * HD;
  float* score = z + (size_t)N * NUM_HEADS;

  zero_f32<<<1024, 256, 0, stream>>>(z,   (long)N * NUM_HEADS);
  zero_f32<<<4096, 256, 0, stream>>>(out, (long)N * HD);

  dim3 ggrid((N + ROWS_PB - 1) / ROWS_PB, 3);
  qkv_proj_wmma<<<ggrid, 256, 0, stream>>>(h, N, Wq, bq, Q, Wk, bk, K, Wv, bv, V);

  edge_scores<<<(E + 7) / 8, 256, 0, stream>>>(Q, K, src, dst, score, z, E);

  aggregate<<<E, 256, 0, stream>>>(V, score, z, src, dst, out, E);
}